// DepthAnythingV2Wrapper_30605936951816
// MI455X (gfx1250) — compile-verified
//
#include <hip/hip_runtime.h>
#include <stdint.h>

namespace {
constexpr int H = 2160;
constexpr int W = 3840;
constexpr int HW = H * W;
constexpr int ROW0    = 1080;   // H_START
constexpr int ROW_END = 2159;   // H_END  (exclusive)
constexpr int COL0    = 1;      // W_START
constexpr int COL_END = 3839;   // W_END  (exclusive)

constexpr int TX  = 64;         // output cols per block
constexpr int TYB = 32;         // output rows per block
constexpr int LH  = TYB + 2;    // 34 staged rows (halo)
constexpr int LW  = TX + 2;     // 66 staged cols (halo)
constexpr int LP  = 68;         // LDS row pitch (68 % 64 = 4 -> conflict-free)
constexpr int STAGE_N = LH * LW; // 2244

// u[j] = -span + j*step ; uv_scaled = u * (W / BEV_WIDTH_M)
constexpr float SPANF = 0.87134857056f;            // 16/sqrt(337)*3839/3840
constexpr float STEPF = (float)(2.0 * 0.87134857056 / 3839.0);
constexpr float UVSCALE = 192.0f;   // W / 20
constexpr float XOFF    = 1920.0f;  // W / 2
constexpr float ZSCALE  = 108.0f;   // H / 20
}

typedef __attribute__((ext_vector_type(2))) float v2f;
typedef __attribute__((ext_vector_type(4))) float v4f;
typedef __attribute__((ext_vector_type(8))) float v8f;

// CDNA5 async global->LDS copy (ASYNCcnt path, CDNA5 ISA).
__device__ __forceinline__ void async_ld_b32(const float* gp, unsigned lds_off) {
  asm volatile("global_load_async_to_lds_b32 %0, %1, off"
               :: "v"(lds_off), "v"((unsigned long long)(uintptr_t)gp)
               : "memory");
}
__device__ __forceinline__ void wait_async0() {
  asm volatile("s_wait_asynccnt 0" ::: "memory");
}

// Kernel 1: copy depth out (nontemporal) + zero BEV half (RT store: keep the
// lines in the 192MB L2 so the scatter hits L2). Warms depth into L2.
__global__ void __launch_bounds__(256)
init_out_kernel(const float* __restrict__ depth, float* __restrict__ out) {
  const size_t i = (size_t)blockIdx.x * 256 + threadIdx.x;   // float4 index
  const v4f v = ((const v4f*)depth)[i];
  __builtin_nontemporal_store(v, ((v4f*)out) + i);
  ((v4f*)(out + HW))[i] = (v4f){0.f, 0.f, 0.f, 0.f};
}

// Kernel 2: async-stage tile -> vertical 3-tap (VALU) -> horizontal 3-tap as
// banded WMMA (v_wmma_f32_16x16x4_f32, 5 K-chunks) -> threshold + BEV scatter.
__global__ void __launch_bounds__(256)
sobel_bev_scatter(const float* __restrict__ depth,
                  const float* __restrict__ thr_p,
                  float* __restrict__ bev) {
  __shared__ __align__(16) float T [LH][LP];   // staged depth tile + halo
  __shared__ __align__(16) float Vx[TYB][LP];  // vertical [1,2,1] (for Gx)
  __shared__ __align__(16) float Vy[TYB][LP];  // vertical [-1,0,1] (for Gy)

  const int tid  = threadIdx.x;        // 0..255, waves = contiguous 32
  const int lane = tid & 31;
  const int wv   = tid >> 5;           // 0..7

  const int r0 = ROW0 + blockIdx.y * TYB - 1;  // staged top row (>= 1079)
  const int c0 = COL0 + blockIdx.x * TX  - 1;  // staged left col (>= 0)

  // ---- Phase A: async global->LDS staging of the halo tile ----
  for (int i = tid; i < STAGE_N; i += 256) {
    const int r = i / LW;
    const int c = i - r * LW;
    const int gr = min(r0 + r, H - 1);         // clamp only on edge tiles
    const int gc = min(c0 + c, W - 1);
    async_ld_b32(depth + (size_t)gr * W + gc, (unsigned)(uintptr_t)&T[r][c]);
  }
  wait_async0();
  __syncthreads();

  // ---- Phase B: vertical 3-tap into LDS; zero the 2 pad columns so the
  //      zero-coefficient padded-K WMMA chunks never touch garbage ----
  for (int i = tid; i < TYB * LP; i += 256) {
    const int r = i / LP;
    const int c = i - r * LP;
    float vx = 0.f, vy = 0.f;
    if (c < LW) {
      const float t0 = T[r][c], t1 = T[r + 1][c], t2 = T[r + 2][c];
      vx = t0 + 2.0f * t1 + t2;   // smooth  (feeds Gx)
      vy = t2 - t0;               // diff    (feeds Gy)
    }
    Vx[r][c] = vx;
    Vy[r][c] = vy;
  }
  __syncthreads();

  // ---- Phase C: horizontal 3-tap as banded matmul, one 16x16 sub-tile per
  //      wave.  G(16x16) = V(16x20 padded) x Band(20x16), 5 chunks of K=4.
  const int sx = wv & 3;          // sub-tile col group (0..3)
  const int sy = wv >> 2;         // sub-tile row group (0..1)
  const int mr = sy * 16;         // sub-tile row base in Vx/Vy
  const int nc = sx * 16;         // sub-tile col base
  const int m     = lane & 15;    // A-layout: M = lane%16
  const int khalf = (lane >> 4) << 1;  // A/B-layout K offset for upper lanes
  const int n     = lane & 15;    // B/D-layout: N = lane%16

  // Banded B constants, same for every sub-tile: B[kk][j] = coef[kk - j].
  // B layout: VGPR v, lanes0-15 -> K=v, lanes16-31 -> K=v+2.
  v2f bd[5], bs[5];
#pragma unroll
  for (int kc = 0; kc < 5; ++kc) {
#pragma unroll
    for (int v = 0; v < 2; ++v) {
      const int t = kc * 4 + v + khalf - n;         // kk - j
      bd[kc][v] = (t == 0) ? -1.0f : ((t == 2) ? 1.0f : 0.0f);      // [-1,0,1]
      bs[kc][v] = (t == 1) ? 2.0f : (((t == 0) | (t == 2)) ? 1.0f : 0.0f); // [1,2,1]
    }
  }

  v8f gx = {}, gy = {};
#pragma unroll
  for (int kc = 0; kc < 5; ++kc) {
    const int col = nc + kc * 4 + khalf;            // even -> 8B aligned
    const v2f ax = *(const v2f*)&Vx[mr + m][col];   // A chunk: ds_load_b64
    const v2f ay = *(const v2f*)&Vy[mr + m][col];
    gx = __builtin_amdgcn_wmma_f32_16x16x4_f32(false, ax, false, bd[kc],
                                               (short)0, gx, false, false);
    gy = __builtin_amdgcn_wmma_f32_16x16x4_f32(false, ay, false, bs[kc],
                                               (short)0, gy, false, false);
  }

  // ---- Phase D: threshold + scatter straight from the D registers.
  // D layout: VGPR v, lanes0-15 -> M=v, lanes16-31 -> M=v+8.
  const float thr = *thr_p;
  const int mbase = (lane >> 4) << 3;
  const int ocol  = COL0 + blockIdx.x * TX + nc + n;
  const float u192 = fmaf((float)ocol, STEPF, -SPANF) * UVSCALE;

  if (ocol < COL_END) {
#pragma unroll
    for (int v = 0; v < 8; ++v) {
      const int M    = mbase + v;
      const int orow = ROW0 + blockIdx.y * TYB + mr + M;
      if (orow < ROW_END) {
        const float gxv = gx[v], gyv = gy[v];
        const float mag = sqrtf(gxv * gxv + gyv * gyv);
        if (mag > thr) {
          const float e = T[mr + M + 1][nc + n + 1];   // center depth
          int x = (int)(u192 * e + XOFF);   // trunc == astype(int32)
          x = min(max(x, 0), W - 1);
          int z = (int)(e * ZSCALE);
          z = min(max(z, 0), H - 1);
          bev[(size_t)(H - 1 - z) * W + x] = 1.0f;     // flip folded in
        }
      }
    }
  }
}

extern "C" void kernel_launch(void* const* d_in, const int* in_sizes, int n_in,
                              void* d_out, int out_size, void* d_ws, size_t ws_size,
                              hipStream_t stream) {
  (void)in_sizes; (void)n_in; (void)out_size; (void)d_ws; (void)ws_size;
  const float* depth = (const float*)d_in[0];
  const float* thr   = (const float*)d_in[1];
  float* out = (float*)d_out;

  init_out_kernel<<<dim3(HW / 4 / 256), dim3(256), 0, stream>>>(depth, out);

  dim3 blk(256);
  dim3 grd((COL_END - COL0 + TX - 1) / TX,    // 60
           (ROW_END - ROW0 + TYB - 1) / TYB); // 34
  sobel_bev_scatter<<<grd, blk, 0, stream>>>(depth, thr, out + HW);
}